// InvariantPointAttention_84121229459832
// MI455X (gfx1250) — compile-verified
//
#include <hip/hip_runtime.h>
#include <hip/hip_bf16.h>
#include <math.h>

namespace {

constexpr int kN    = 768;
constexpr int kCS   = 384;
constexpr int kCZ   = 128;
constexpr int kCH   = 16;
constexpr int kH    = 12;
constexpr int kPQ   = 4;
constexpr int kPV   = 8;
constexpr int kLinW = 1152;   // 192(q) + 384(kv) + 144(qp) + 432(kvp)
constexpr int kOcatW= 2112;   // 192 + 3*96 + 96 + 1536
constexpr float kInf = 100000.0f;
constexpr float kEps = 1e-8f;

typedef __attribute__((ext_vector_type(2))) float v2f;
typedef __attribute__((ext_vector_type(8))) float v8f;
typedef __attribute__((ext_vector_type(4))) unsigned int v4u;
typedef __attribute__((ext_vector_type(8))) int v8i;
typedef __attribute__((ext_vector_type(4))) int v4i;

// Full-precision fp32 WMMA: D(16x16) = A(16x4) * B(4x16) + C
__device__ __forceinline__ v8f wmma4(v2f a, v2f b, v8f c) {
  return __builtin_amdgcn_wmma_f32_16x16x4_f32(false, a, false, b, (short)0, c,
                                               false, false);
}

// ---------------------------------------------------------------------------
// TDM: DMA a 128x128 f32 row-major tile (contiguous rows, stride 128) from
// global memory into LDS.  D# packed per CDNA5 ISA.
// 6-arg builtin form (this toolchain): (g0, g1, g2, g3, g4, cpol).
// ---------------------------------------------------------------------------
__device__ __forceinline__ void tdm_load_tile_128x128(const float* gsrc,
                                                      unsigned lds_byte_addr) {
  const unsigned long long ga = (unsigned long long)(size_t)gsrc;
  v4u g0;
  g0[0] = 1u;                                           // count=1, user mode
  g0[1] = lds_byte_addr;                                // lds_addr
  g0[2] = (unsigned)(ga & 0xFFFFFFFFull);               // global_addr[31:0]
  g0[3] = (unsigned)((ga >> 32) & 0x01FFFFFFull)        // global_addr[56:32]
        | (2u << 30);                                   // type=2 ("image")
  v8i g1;
  g1[0] = (int)(2u << 16);        // workgroup_mask=0, data_size=2 (4 bytes)
  g1[1] = (int)(128u << 16);      // tensor_dim0[15:0]=128 in bits[63:48]
  g1[2] = (int)(128u << 16);      // tensor_dim0 hi=0 | tensor_dim1[15:0]=128
  g1[3] = (int)(128u << 16);      // tensor_dim1 hi=0 | tile_dim0=128
  g1[4] = 128;                    // tile_dim1=128, tile_dim2=0
  g1[5] = 128;                    // tensor_dim0_stride[31:0]=128
  g1[6] = 0;                      // stride0 hi | stride1 lo
  g1[7] = 0;                      // stride1 hi
  v4i gz4 = {};
  v8i gz8 = {};
  __builtin_amdgcn_tensor_load_to_lds(g0, g1, gz4, gz4, gz8, 0);
}

// ---------------------------------------------------------------------------
// K1: fused input projections: lin[n][0:1152] = s @ [w_q|w_kv|w_qp|w_kvp]^T + b
// ---------------------------------------------------------------------------
__global__ void k_proj(const float* __restrict__ s,
                       const float* __restrict__ wq, const float* __restrict__ bq,
                       const float* __restrict__ wkv, const float* __restrict__ bkv,
                       const float* __restrict__ wqp, const float* __restrict__ bqp,
                       const float* __restrict__ wkvp, const float* __restrict__ bkvp,
                       float* __restrict__ lin) {
  const int wave = threadIdx.x >> 5;
  const int lane = threadIdx.x & 31;
  const int j = lane & 15, half = lane >> 4;
  const int rowBase = blockIdx.x * 16;
  const int colBase = (blockIdx.y * 4 + wave) * 16;

  const float* w; const float* bias; int cl;
  if (colBase < 192)      { w = wq;   bias = bq;   cl = colBase; }
  else if (colBase < 576) { w = wkv;  bias = bkv;  cl = colBase - 192; }
  else if (colBase < 720) { w = wqp;  bias = bqp;  cl = colBase - 576; }
  else                    { w = wkvp; bias = bkvp; cl = colBase - 720; }

  const float* arow = s + (size_t)(rowBase + j) * kCS;
  const float* brow = w + (size_t)(cl + j) * kCS;
  v8f acc = {};
  for (int k0 = 0; k0 < kCS; k0 += 4) {
    v2f a, b;
    a.x = arow[k0 + 2 * half]; a.y = arow[k0 + 2 * half + 1];
    b.x = brow[k0 + 2 * half]; b.y = brow[k0 + 2 * half + 1];
    acc = wmma4(a, b, acc);
  }
  const float bv = bias[cl + j];
#pragma unroll
  for (int r = 0; r < 8; r++)
    lin[(size_t)(rowBase + r + 8 * half) * kLinW + colBase + j] = acc[r] + bv;
}

// ---------------------------------------------------------------------------
// K1b: rotate/translate points into global frame
// ---------------------------------------------------------------------------
__global__ void k_rotate(const float* __restrict__ lin, const float* __restrict__ rot,
                         const float* __restrict__ trans,
                         float* __restrict__ qpts, float* __restrict__ kpts,
                         float* __restrict__ vpts) {
  const int idx = blockIdx.x * blockDim.x + threadIdx.x;
  if (idx >= kN * 192) return;
  const int n = idx / 192;
  const int p = idx % 192;
  const float* R = rot + (size_t)n * 9;
  const float* T = trans + (size_t)n * 3;
  float x[3], y[3];
  if (p < 48) {
    for (int i = 0; i < 3; i++) x[i] = lin[(size_t)n * kLinW + 576 + i * 48 + p];
    for (int i = 0; i < 3; i++)
      y[i] = R[i*3+0]*x[0] + R[i*3+1]*x[1] + R[i*3+2]*x[2] + T[i];
    const int h = p >> 2, pp = p & 3;
    float* dst = qpts + (((size_t)n * kH + h) * kPQ + pp) * 3;
    dst[0] = y[0]; dst[1] = y[1]; dst[2] = y[2];
  } else {
    const int pk = p - 48;
    for (int i = 0; i < 3; i++) x[i] = lin[(size_t)n * kLinW + 720 + i * 144 + pk];
    for (int i = 0; i < 3; i++)
      y[i] = R[i*3+0]*x[0] + R[i*3+1]*x[1] + R[i*3+2]*x[2] + T[i];
    const int h = pk / 12, r = pk % 12;
    float* dst = (r < kPQ) ? (kpts + (((size_t)n * kH + h) * kPQ + r) * 3)
                           : (vpts + (((size_t)n * kH + h) * kPV + (r - kPQ)) * 3);
    dst[0] = y[0]; dst[1] = y[1]; dst[2] = y[2];
  }
}

// ---------------------------------------------------------------------------
// K2: bmat[h][n][m] = sqrt(1/3)*(z[n][m][:].w_b[h] + b_b[h])
// z[n] streamed through LDS by TDM in 128-row double-buffered tiles.
// ---------------------------------------------------------------------------
__global__ void __launch_bounds__(256) k_bmat(const float* __restrict__ z,
                                              const float* __restrict__ wb,
                                              const float* __restrict__ bb,
                                              float* __restrict__ bmat) {
  __shared__ float zbuf[2][128 * kCZ];            // 2 x 64 KB
  const int wave = threadIdx.x >> 5, lane = threadIdx.x & 31;
  const int j = lane & 15, half = lane >> 4;
  const int n = blockIdx.x;
  const float* zsrc = z + (size_t)n * kN * kCZ;
  const float* brow = wb + (size_t)((j < kH) ? j : 0) * kCZ;
  const float c3 = 0.5773502691896258f;           // sqrt(1/3)

  if (wave == 0)
    tdm_load_tile_128x128(zsrc, (unsigned)(size_t)&zbuf[0][0]);

  for (int c = 0; c < 6; c++) {
    if (wave == 0) {
      if (c < 5) {
        tdm_load_tile_128x128(zsrc + (size_t)(c + 1) * 128 * kCZ,
                              (unsigned)(size_t)&zbuf[(c + 1) & 1][0]);
        __builtin_amdgcn_s_wait_tensorcnt(1);     // chunk c landed
      } else {
        __builtin_amdgcn_s_wait_tensorcnt(0);
      }
    }
    __syncthreads();
    const float* arow = &zbuf[c & 1][(size_t)(wave * 16 + j) * kCZ];  // A[M=m][K=c]
    v8f acc = {};
    for (int k0 = 0; k0 < kCZ; k0 += 4) {
      v2f a, b;
      a.x = arow[k0 + 2 * half]; a.y = arow[k0 + 2 * half + 1];
      b.x = brow[k0 + 2 * half]; b.y = brow[k0 + 2 * half + 1];
      acc = wmma4(a, b, acc);
    }
    if (j < kH) {
      const float bias = bb[j];
      const int mBase = c * 128 + wave * 16;
#pragma unroll
      for (int r = 0; r < 8; r++)
        bmat[((size_t)j * kN + n) * kN + mBase + r + 8 * half] = c3 * (acc[r] + bias);
    }
    __syncthreads();                              // done reading buf[c&1]
  }
}

// ---------------------------------------------------------------------------
// K3: attention logits (K=28 fused q.k + point cross-term) + softmax per
// (head, 16-row n-tile); probs written in place over bmat.
// ---------------------------------------------------------------------------
__global__ void k_attn(const float* __restrict__ lin,
                       const float* __restrict__ qpts, const float* __restrict__ kpts,
                       const float* __restrict__ mask,
                       const float* __restrict__ head_weights,
                       float* __restrict__ attn) {
  __shared__ float lds[16 * kN];
  __shared__ float qterm_s[16];
  __shared__ float maskn_s[16];
  const int tid = threadIdx.x;
  const int wave = tid >> 5, lane = tid & 31;
  const int j = lane & 15, half = lane >> 4;
  const int h = blockIdx.x;
  const int nBase = blockIdx.y * 16;

  const float c1 = 0.14433756729740643f;  // sqrt(1/(3*C_H))
  const float hw = logf(1.0f + __expf(head_weights[h])) * 0.13608276348795434f;

  if (tid < 16) {
    const int n = nBase + tid;
    const float* qp = qpts + ((size_t)n * kH + h) * 12;
    float q2 = 0.f;
#pragma unroll
    for (int c = 0; c < 12; c++) { float t = qp[c]; q2 += t * t; }
    qterm_s[tid] = -0.5f * hw * q2;
    maskn_s[tid] = mask[n];
  }
  __syncthreads();

  v2f areg[7];
  {
    const int n = nBase + j;
    const float* qrow = lin + (size_t)n * kLinW + h * kCH;
    const float* qprow = qpts + ((size_t)n * kH + h) * 12;
#pragma unroll
    for (int kk = 0; kk < 7; kk++) {
#pragma unroll
      for (int r = 0; r < 2; r++) {
        const int kq = 4 * kk + 2 * half + r;
        const float v = (kq < 16) ? (c1 * qrow[kq]) : (hw * qprow[kq - 16]);
        if (r == 0) areg[kk].x = v; else areg[kk].y = v;
      }
    }
  }

  for (int tt = 0; tt < 6; tt++) {
    const int mBase = (wave * 6 + tt) * 16;
    const int m = mBase + j;
    const float* krow  = lin + (size_t)m * kLinW + 192 + h * 32;
    const float* kprow = kpts + ((size_t)m * kH + h) * 12;
    v8f acc = {};
#pragma unroll
    for (int kk = 0; kk < 7; kk++) {
      const int kq0 = 4 * kk + 2 * half;
      v2f b;
      b.x = (kq0     < 16) ? krow[kq0]     : kprow[kq0 - 16];
      b.y = (kq0 + 1 < 16) ? krow[kq0 + 1] : kprow[kq0 - 15];
      acc = wmma4(areg[kk], b, acc);
    }
    float k2 = 0.f;
#pragma unroll
    for (int c = 0; c < 12; c++) { float t = kprow[c]; k2 += t * t; }
    const float kterm = -0.5f * hw * k2;
    const float maskm = mask[m];
    const float* bmrow = attn + ((size_t)h * kN + nBase) * kN + m;
#pragma unroll
    for (int r = 0; r < 8; r++) {
      const int rloc = r + 8 * half;
      lds[rloc * kN + m] = acc[r] + bmrow[(size_t)rloc * kN] + qterm_s[rloc] +
                           kterm + (maskn_s[rloc] * maskm - 1.0f) * kInf;
    }
  }
  __syncthreads();

  for (int rr = 0; rr < 2; rr++) {
    const int row = wave * 2 + rr;
    float mx = -3.4e38f;
    for (int c = lane; c < kN; c += 32) mx = fmaxf(mx, lds[row * kN + c]);
#pragma unroll
    for (int off = 16; off > 0; off >>= 1) mx = fmaxf(mx, __shfl_xor(mx, off, 32));
    float sm = 0.f;
    for (int c = lane; c < kN; c += 32) sm += __expf(lds[row * kN + c] - mx);
#pragma unroll
    for (int off = 16; off > 0; off >>= 1) sm += __shfl_xor(sm, off, 32);
    const float inv = 1.0f / sm;
    float* orow = attn + ((size_t)h * kN + nBase + row) * kN;
    for (int c = lane; c < kN; c += 32)
      orow[c] = __expf(lds[row * kN + c] - mx) * inv;
  }
}

// ---------------------------------------------------------------------------
// K4: fused o = a@v and o_pt_global = a@v_pts per head: one A-load feeds
// three accumulators (v | pts[0:16] | pts[16:24]) -> attn read exactly once.
// ---------------------------------------------------------------------------
__global__ void k_ov(const float* __restrict__ attn, const float* __restrict__ lin,
                     const float* __restrict__ vpts,
                     float* __restrict__ ocat, float* __restrict__ optg) {
  const int lane = threadIdx.x & 31;
  const int j = lane & 15, half = lane >> 4;
  const int nBase = blockIdx.x * 16;
  const int h = blockIdx.y;
  const float* arow = attn + ((size_t)h * kN + nBase + j) * kN;
  const int c2a = j;            // point flat col, tile 1
  const int c2b = 16 + j;       // point flat col, tile 2 (valid < 24)
  const bool vb = (c2b < 24);
  v8f acc0 = {}, acc1 = {}, acc2 = {};
  for (int k0 = 0; k0 < kN; k0 += 4) {
    const int m0 = k0 + 2 * half;
    v2f a; a.x = arow[m0]; a.y = arow[m0 + 1];
    const float* l0 = lin + (size_t)m0 * kLinW + 192 + h * 32 + 16 + j;
    v2f b0; b0.x = l0[0]; b0.y = l0[kLinW];
    const float* p0 = vpts + ((size_t)m0 * kH + h) * 24;
    v2f b1; b1.x = p0[c2a]; b1.y = p0[kH * 24 + c2a];
    v2f b2; b2.x = vb ? p0[c2b] : 0.f; b2.y = vb ? p0[kH * 24 + c2b] : 0.f;
    acc0 = wmma4(a, b0, acc0);
    acc1 = wmma4(a, b1, acc1);
    acc2 = wmma4(a, b2, acc2);
  }
#pragma unroll
  for (int r = 0; r < 8; r++) {
    const int n = nBase + r + 8 * half;
    ocat[(size_t)n * kOcatW + h * kCH + j] = acc0[r];
    optg[((size_t)n * kH + h) * 24 + c2a] = acc1[r];
    if (vb) optg[((size_t)n * kH + h) * 24 + c2b] = acc2[r];
  }
}

// ---------------------------------------------------------------------------
// K5: inverse-frame rotation of o_pt + norms
// ---------------------------------------------------------------------------
__global__ void k_optfin(const float* __restrict__ optg, const float* __restrict__ rot,
                         const float* __restrict__ trans, float* __restrict__ ocat) {
  const int idx = blockIdx.x * blockDim.x + threadIdx.x;
  if (idx >= kN * kH * kPV) return;
  const int n  = idx / (kH * kPV);
  const int hp = idx % (kH * kPV);
  const float* R = rot + (size_t)n * 9;
  const float* T = trans + (size_t)n * 3;
  const float* g = optg + (size_t)n * (kH * kPV * 3) + hp * 3;
  const float g0 = g[0] - T[0], g1 = g[1] - T[1], g2 = g[2] - T[2];
  float l[3];
#pragma unroll
  for (int i = 0; i < 3; i++)
    l[i] = R[0 * 3 + i] * g0 + R[1 * 3 + i] * g1 + R[2 * 3 + i] * g2;
  const float nrm = sqrtf(l[0]*l[0] + l[1]*l[1] + l[2]*l[2] + kEps);
  float* o = ocat + (size_t)n * kOcatW;
  o[192 + 0 * 96 + hp] = l[0];
  o[192 + 1 * 96 + hp] = l[1];
  o[192 + 2 * 96 + hp] = l[2];
  o[480 + hp] = nrm;
}

// ---------------------------------------------------------------------------
// K6: o_pair[n][h][c] = sum_m attn[h][n][m] * z[n][m][c]
// z[n] streamed via TDM double-buffered 128x128 LDS tiles; 8 waves own the
// 8 column tiles of 128; K accumulated across the 6 chunks.
// ---------------------------------------------------------------------------
__global__ void __launch_bounds__(256) k_opair(const float* __restrict__ attn,
                                               const float* __restrict__ z,
                                               float* __restrict__ ocat) {
  __shared__ float zbuf[2][128 * kCZ];            // 2 x 64 KB
  const int wave = threadIdx.x >> 5, lane = threadIdx.x & 31;
  const int j = lane & 15, half = lane >> 4;
  const int n = blockIdx.x;
  const int ct = wave;                            // column tile 0..7
  const int hrow = (j < kH) ? j : 0;              // pad head rows
  const float* arow = attn + ((size_t)hrow * kN + n) * kN;   // A[M=h][K=m]
  const float* zsrc = z + (size_t)n * kN * kCZ;

  if (wave == 0)
    tdm_load_tile_128x128(zsrc, (unsigned)(size_t)&zbuf[0][0]);

  v8f acc = {};
  for (int c = 0; c < 6; c++) {
    if (wave == 0) {
      if (c < 5) {
        tdm_load_tile_128x128(zsrc + (size_t)(c + 1) * 128 * kCZ,
                              (unsigned)(size_t)&zbuf[(c + 1) & 1][0]);
        __builtin_amdgcn_s_wait_tensorcnt(1);
      } else {
        __builtin_amdgcn_s_wait_tensorcnt(0);
      }
    }
    __syncthreads();
    const float* zb = &zbuf[c & 1][0];
    const float* ar = arow + c * 128;
    for (int k0 = 0; k0 < 128; k0 += 4) {
      const int m0 = k0 + 2 * half;
      v2f a, b;
      a.x = ar[m0]; a.y = ar[m0 + 1];
      b.x = zb[(size_t)m0 * kCZ + ct * 16 + j];
      b.y = zb[(size_t)(m0 + 1) * kCZ + ct * 16 + j];
      acc = wmma4(a, b, acc);
    }
    __syncthreads();
  }
#pragma unroll
  for (int r = 0; r < 8; r++) {
    const int hd = r + 8 * half;
    if (hd < kH)
      ocat[(size_t)n * kOcatW + 576 + hd * kCZ + ct * 16 + j] = acc[r];
  }
}

// ---------------------------------------------------------------------------
// K7: final projection: out = o_cat (768x2112) @ w_out^T + b_out
// ---------------------------------------------------------------------------
__global__ void k_outproj(const float* __restrict__ ocat, const float* __restrict__ wout,
                          const float* __restrict__ bout, float* __restrict__ out) {
  const int wave = threadIdx.x >> 5, lane = threadIdx.x & 31;
  const int j = lane & 15, half = lane >> 4;
  const int nBase = blockIdx.x * 16;
  const int cBase = (blockIdx.y * 4 + wave) * 16;
  const float* arow = ocat + (size_t)(nBase + j) * kOcatW;
  const float* brow = wout + (size_t)(cBase + j) * kOcatW;
  v8f acc = {};
#pragma unroll 4
  for (int k0 = 0; k0 < kOcatW; k0 += 4) {
    v2f a, b;
    a.x = arow[k0 + 2 * half]; a.y = arow[k0 + 2 * half + 1];
    b.x = brow[k0 + 2 * half]; b.y = brow[k0 + 2 * half + 1];
    acc = wmma4(a, b, acc);
  }
  const float bv = bout[cBase + j];
#pragma unroll
  for (int r = 0; r < 8; r++)
    out[(size_t)(nBase + r + 8 * half) * kCS + cBase + j] = acc[r] + bv;
}

}  // namespace

extern "C" void kernel_launch(void* const* d_in, const int* in_sizes, int n_in,
                              void* d_out, int out_size, void* d_ws, size_t ws_size,
                              hipStream_t stream) {
  const float* s     = (const float*)d_in[0];
  const float* z     = (const float*)d_in[1];
  const float* rot   = (const float*)d_in[2];
  const float* trans = (const float*)d_in[3];
  const float* mask  = (const float*)d_in[4];
  const float* w_q   = (const float*)d_in[5];
  const float* b_q   = (const float*)d_in[6];
  const float* w_kv  = (const float*)d_in[7];
  const float* b_kv  = (const float*)d_in[8];
  const float* w_qp  = (const float*)d_in[9];
  const float* b_qp  = (const float*)d_in[10];
  const float* w_kvp = (const float*)d_in[11];
  const float* b_kvp = (const float*)d_in[12];
  const float* w_b   = (const float*)d_in[13];
  const float* b_b   = (const float*)d_in[14];
  const float* hwts  = (const float*)d_in[15];
  const float* w_out = (const float*)d_in[16];
  const float* b_out = (const float*)d_in[17];
  float* out = (float*)d_out;

  float* ws = (float*)d_ws;
  size_t off = 0;
  float* lin  = ws + off; off += (size_t)kN * kLinW;
  float* qpts = ws + off; off += (size_t)kN * kH * kPQ * 3;
  float* kpts = ws + off; off += (size_t)kN * kH * kPQ * 3;
  float* vpts = ws + off; off += (size_t)kN * kH * kPV * 3;
  float* attn = ws + off; off += (size_t)kH * kN * kN;   // bmat -> probs in place
  float* optg = ws + off; off += (size_t)kN * kH * kPV * 3;
  float* ocat = ws + off;

  k_proj<<<dim3(kN / 16, (kLinW / 16) / 4), 128, 0, stream>>>(
      s, w_q, b_q, w_kv, b_kv, w_qp, b_qp, w_kvp, b_kvp, lin);
  k_rotate<<<(kN * 192 + 255) / 256, 256, 0, stream>>>(lin, rot, trans, qpts, kpts, vpts);
  k_bmat<<<dim3(kN), 256, 0, stream>>>(z, w_b, b_b, attn);
  k_attn<<<dim3(kH, kN / 16), 256, 0, stream>>>(lin, qpts, kpts, mask, hwts, attn);
  k_ov<<<dim3(kN / 16, kH), 32, 0, stream>>>(attn, lin, vpts, ocat, optg);
  k_optfin<<<(kN * kH * kPV + 255) / 256, 256, 0, stream>>>(optg, rot, trans, ocat);
  k_opair<<<dim3(kN), 256, 0, stream>>>(attn, z, ocat);
  k_outproj<<<dim3(kN / 16, (kCS / 16) / 4), 128, 0, stream>>>(ocat, w_out, b_out, out);
}